// CritiGraph_9251359555762
// MI455X (gfx1250) — compile-verified
//
#include <hip/hip_runtime.h>
#include <stdint.h>
#include <stddef.h>

namespace {
constexpr int cH  = 16;          // bits per code
constexpr int cTP = 8;           // parallel copies
constexpr int cK  = 16;          // masks per bit
constexpr int cB  = 4;
constexpr int cT  = 32;
constexpr int cHK = cH * cK;     // 256
constexpr int cM  = 2 * cHK + 1; // 513 candidates
}

#if defined(__HIP_DEVICE_COMPILE__) && \
    __has_builtin(__builtin_amdgcn_global_load_async_to_lds_b128) && \
    __has_builtin(__builtin_amdgcn_s_wait_asynccnt)
#define USE_ASYNC_LDS 1
#else
#define USE_ASYNC_LDS 0
#endif

typedef int v4i __attribute__((ext_vector_type(4)));
typedef __attribute__((address_space(1))) v4i as1_v4i;   // global int4
typedef __attribute__((address_space(3))) v4i as3_v4i;   // LDS int4

__global__ __launch_bounds__(256) void critigraph_kernel(
    const int*   __restrict__ sta_ind,    // (B,T)
    const float* __restrict__ logits,     // (B,T,T)
    const int*   __restrict__ locations,  // (N,TP)
    const int*   __restrict__ rmasks,     // (B*T,H,K,TP)
    float*       __restrict__ out)        // (B,T,TP)
{
    __shared__ int   s_abs[cT * cTP];     // |sta_loc[b][j][tp]|
    __shared__ float s_sgn[cT * cTP];     // sign(sta_loc) as float
    __shared__ float s_A[cT * cTP];       // (dsps - dsp)/TP - logits
    __shared__ float s_red[cT * cTP];     // dsp scratch, then min-reduce
    __shared__ int   s_rm[cHK * cTP];     // 8 KB mask tile for this (b,i)

    const int t    = threadIdx.x;
    const int tp   = t & 7;               // 0..7  : TP lane
    const int slot = t >> 3;              // 0..31 : j / candidate stride
    const int bi   = blockIdx.x;          // b*T + i
    const int b    = bi >> 5;
    const int i    = bi & 31;

    const int* rm_blk = rmasks + (size_t)bi * (cHK * cTP);

    // ---- Phase 0: stage 8 KB of random_masks into LDS (async, overlapped) ----
    // 512 x 16B chunks; each of the 256 threads issues two b128 async copies.
#if USE_ASYNC_LDS
    {
        as1_v4i* g = (as1_v4i*)rm_blk;
        as3_v4i* l = (as3_v4i*)s_rm;
        __builtin_amdgcn_global_load_async_to_lds_b128(g + t,       l + t,       0, 0);
        __builtin_amdgcn_global_load_async_to_lds_b128(g + 256 + t, l + 256 + t, 0, 0);
    }
#else
    for (int c = t; c < cHK * cTP; c += 256) s_rm[c] = rm_blk[c];
#endif

    // ---- Phase 1: gather sta_loc[b, slot, tp]; store |v| and sign ----
    {
        int idx = sta_ind[b * cT + slot];
        int v   = locations[(size_t)idx * cTP + tp];
        s_abs[slot * 8 + tp] = (v < 0) ? -v : v;
        s_sgn[slot * 8 + tp] = (v > 0) ? 1.0f : ((v < 0) ? -1.0f : 0.0f);
    }
    __syncthreads();

    // 1 - table[x] == (clz32(x+1) - 16) / 16   for x in [0, 65535]
    const int   absi = s_abs[i * 8 + tp];
    const float sgni = s_sgn[i * 8 + tp];

    // ---- Phase 2a: dsp(i, j=slot, tp) ----
    {
        int   xr = absi ^ s_abs[slot * 8 + tp];
        float d  = sgni * s_sgn[slot * 8 + tp] *
                   (float)(__builtin_clz((unsigned)(xr + 1)) - 16) * 0.0625f;
        s_red[slot * 8 + tp] = d;
    }
    __syncthreads();

    // ---- Phase 2b: dsps(i,j) and affine constant A(j,tp) ----
    {
        float dsps = 0.0f;
#pragma unroll
        for (int q = 0; q < 8; ++q) dsps += s_red[slot * 8 + q];
        float a = (dsps - s_red[slot * 8 + tp]) * 0.125f - logits[bi * cT + slot];
        s_A[slot * 8 + tp] = a;
    }
#if USE_ASYNC_LDS
    __builtin_amdgcn_s_wait_asynccnt(0);
#endif
    __syncthreads();

    // Candidate generator: res = (absi ^ (1<<h)) ^ mask[h,k,tp]; m = h*K+k;
    // cnc = [res (m<256), absi (m==256), -res (m>256)].
    // sv carries sign(c2) folded with the 1/(16*TP)=1/128 distance scale.
    auto cand = [&](int mm, int& r, float& sv) {
        if (mm == 256) {
            r  = absi;
            sv = (r != 0) ? (1.0f / 128.0f) : 0.0f;
        } else {
            int m2 = (mm < 256) ? mm : (mm - 257);      // m2 = h*16 + k
            int h  = m2 >> 4;
            r      = (absi ^ (1 << h)) ^ s_rm[m2 * 8 + tp];
            float mag = (r != 0) ? (1.0f / 128.0f) : 0.0f;
            sv = (mm < 256) ? mag : -mag;
        }
    };

    // ---- Phase 3: hot loop. 16 candidates per thread (stride 32), blocked 4-wide
    //      so the 3 LDS reads per neighbor j are amortized over 4 candidates. ----
    float bestv = 3.4e38f;
#pragma unroll
    for (int g = 0; g < 4; ++g) {
        int   r[4];
        float sv[4];
        float acc[4] = {0.f, 0.f, 0.f, 0.f};
#pragma unroll
        for (int u = 0; u < 4; ++u) cand(slot + (4 * g + u) * 32, r[u], sv[u]);

#pragma unroll 4
        for (int j = 0; j < 32; ++j) {
            int   aj = s_abs[j * 8 + tp];
            float sj = s_sgn[j * 8 + tp];
            float Aj = s_A[j * 8 + tp];
#pragma unroll
            for (int u = 0; u < 4; ++u) {
                int   xr = aj ^ r[u];
                float c  = (float)(__builtin_clz((unsigned)(xr + 1)) - 16);
                acc[u] += fabsf(fmaf(sv[u], c * sj, Aj));
            }
        }
#pragma unroll
        for (int u = 0; u < 4; ++u)
            bestv = fminf(bestv, acc[u] * (1.0f / 32.0f));
    }

    // Epilogue: candidate mm = 512 (handled by slot 0 only).
    if (slot == 0) {
        int r; float sv;
        cand(512, r, sv);
        float acc = 0.0f;
#pragma unroll 4
        for (int j = 0; j < 32; ++j) {
            int   xr = s_abs[j * 8 + tp] ^ r;
            float c  = (float)(__builtin_clz((unsigned)(xr + 1)) - 16);
            acc += fabsf(fmaf(sv, c * s_sgn[j * 8 + tp], s_A[j * 8 + tp]));
        }
        bestv = fminf(bestv, acc * (1.0f / 32.0f));
    }

    // ---- Phase 4: min-reduce over the 32 candidate slots per tp lane ----
    s_red[t] = bestv;
    __syncthreads();
    for (int s = 16; s > 0; s >>= 1) {
        if (slot < s) {
            float ov = s_red[t + 8 * s];
            if (ov < s_red[t]) s_red[t] = ov;
        }
        __syncthreads();
    }
    if (slot == 0) out[bi * 8 + tp] = s_red[t];
}

extern "C" void kernel_launch(void* const* d_in, const int* in_sizes, int n_in,
                              void* d_out, int out_size, void* d_ws, size_t ws_size,
                              hipStream_t stream) {
    const int*   sta_ind   = (const int*)d_in[0];
    const float* logits    = (const float*)d_in[1];
    const int*   locations = (const int*)d_in[2];
    const int*   rmasks    = (const int*)d_in[3];
    float*       out       = (float*)d_out;
    (void)in_sizes; (void)n_in; (void)out_size; (void)d_ws; (void)ws_size;

    critigraph_kernel<<<dim3(cB * cT), dim3(256), 0, stream>>>(
        sta_ind, logits, locations, rmasks, out);
}